// IMUHOIModel_27745488732349
// MI455X (gfx1250) — compile-verified
//
#include <hip/hip_runtime.h>
#include <stdint.h>

typedef uint32_t u32;
typedef uint64_t u64;
typedef __attribute__((ext_vector_type(4))) unsigned int v4u;
typedef __attribute__((ext_vector_type(8))) int v8i;
typedef __attribute__((ext_vector_type(4))) int v4i;

#define LANES 32
#define CHUNK 8   // timesteps per TDM tile; 4096 % 8 == 0

// ---------------------------------------------------------------------------
// Issue one TDM 2D-tile load: tile_d1 rows x tile_d0 elements (f32), row
// stride row_stride_elems, from gaddr_bytes into LDS at lds_byte_addr.
// Descriptor layout per CDNA5 ISA ch. 8 (D# groups 0..3).
// ---------------------------------------------------------------------------
__device__ __forceinline__ void tdm_load_tile_2d(u32 lds_byte_addr,
                                                 u64 gaddr_bytes,
                                                 u32 tile_d0, u32 tile_d1,
                                                 u32 row_stride_elems,
                                                 u32 tensor_d1_rows) {
    // Group 0 (128b): [1:0]=count=1 (valid, user mode), [63:32]=lds_addr,
    // [120:64]=global_addr, [127:126]=type=2 ("image").
    v4u g0;
    g0.x = 1u;
    g0.y = lds_byte_addr;
    g0.z = (u32)(gaddr_bytes & 0xFFFFFFFFull);
    g0.w = (u32)((gaddr_bytes >> 32) & 0x01FFFFFFull) | (2u << 30);

    // Group 1 (256b): workgroup_mask=0 (not in cluster), data_size=2 (4B),
    // no atomic-barrier / iterate / pad. tensor_dim0 = full row length,
    // tensor_dim1 = rows remaining (OOB rows read as zero), tile dims,
    // tensor_dim0_stride = row stride (elements). tile_dim2 = 0 -> 2D.
    const u32 tensor_d0 = row_stride_elems;
    v8i g1;
    g1[0] = (int)(2u << 16);                                      // data_size=4B
    g1[1] = (int)((tensor_d0 & 0xFFFFu) << 16);                   // dim0[15:0] @ bits63:48
    g1[2] = (int)(((tensor_d0 >> 16) & 0xFFFFu) |
                  ((tensor_d1_rows & 0xFFFFu) << 16));            // dim0[31:16], dim1[15:0]
    g1[3] = (int)(((tensor_d1_rows >> 16) & 0xFFFFu) |
                  ((tile_d0 & 0xFFFFu) << 16));                   // dim1[31:16], tile_dim0
    g1[4] = (int)(tile_d1 & 0xFFFFu);                             // tile_dim1 (tile_dim2=0)
    g1[5] = (int)row_stride_elems;                                // dim0_stride[31:0]
    g1[6] = 0;                                                    // dim0_stride[47:32], dim1_stride lo
    g1[7] = 0;
    // Groups 2/3: dims 2..4 unused (tile_dim2/3/4 = 0).
    v4i g2 = {0, 0, 0, 0};
    v4i g3 = {0, 0, 0, 0};
    // 6-arg toolchain variant: extra int32x8 group (zero-filled) + cpol.
    v8i g4 = {0, 0, 0, 0, 0, 0, 0, 0};
    __builtin_amdgcn_tensor_load_to_lds(g0, g1, g2, g3, g4, 0);
}

// ---------------------------------------------------------------------------
// One wave32 per block; lane = one batch chain. TDM double-buffers the per-
// chunk [32 x CHUNK*{2,6,9}] f32 tiles into LDS while the wave runs the
// sequential contact-FSM / FK recurrence on the previous chunk.
// ---------------------------------------------------------------------------
__global__ __launch_bounds__(LANES, 1)
void fk_obj_trans_kernel(const float* __restrict__ prob,    // [B][T][2]
                         const float* __restrict__ hands,   // [B][T][2][3]
                         const float* __restrict__ rotm,    // [B][T][3][3]
                         const float* __restrict__ tinit,   // [B][3]
                         float* __restrict__ out,           // [B][T][3]
                         int B, int T) {
    __shared__ float s_prob[2][LANES][CHUNK * 2];
    __shared__ float s_hand[2][LANES][CHUNK * 6];
    __shared__ float s_rotm[2][LANES][CHUNK * 9];

    const int lane   = threadIdx.x;
    const int b0     = blockIdx.x * LANES;
    const int b      = b0 + lane;
    const bool live  = (b < B);
    const u32 rows   = (u32)(B - b0);                 // TDM OOB rows -> zero

    // Carried state
    int   cur = -1;
    float px = 0.f, py = 0.f, pz = 0.f;
    if (live) { px = tinit[b*3+0]; py = tinit[b*3+1]; pz = tinit[b*3+2]; }
    float dlx = 0.f, dly = 0.f, dlz = 0.f, dist = 0.f;
    int   prev_lc = 0, prev_rc = 0;

    const int nch = T / CHUNK;

    const u64 prob_base = (u64)(size_t)prob + ((u64)b0 * T * 2) * 4ull;
    const u64 hand_base = (u64)(size_t)hands + ((u64)b0 * T * 6) * 4ull;
    const u64 rotm_base = (u64)(size_t)rotm + ((u64)b0 * T * 9) * 4ull;

    auto issue_chunk = [&](int c, int buf) {
        const u64 t0 = (u64)c * CHUNK;
        tdm_load_tile_2d((u32)(size_t)&s_prob[buf][0][0],
                         prob_base + t0 * 2ull * 4ull,
                         CHUNK * 2, LANES, (u32)(T * 2), rows);
        tdm_load_tile_2d((u32)(size_t)&s_hand[buf][0][0],
                         hand_base + t0 * 6ull * 4ull,
                         CHUNK * 6, LANES, (u32)(T * 6), rows);
        tdm_load_tile_2d((u32)(size_t)&s_rotm[buf][0][0],
                         rotm_base + t0 * 9ull * 4ull,
                         CHUNK * 9, LANES, (u32)(T * 9), rows);
    };

    issue_chunk(0, 0);

    for (int c = 0; c < nch; ++c) {
        const int buf = c & 1;
        if (c + 1 < nch) {
            issue_chunk(c + 1, (c + 1) & 1);
            __builtin_amdgcn_s_wait_tensorcnt(3);   // in-order: chunk c done
        } else {
            __builtin_amdgcn_s_wait_tensorcnt(0);
        }

        #pragma unroll
        for (int tt = 0; tt < CHUNK; ++tt) {
            const int t = c * CHUNK + tt;
            const float p0 = s_prob[buf][lane][tt*2 + 0];
            const float p1 = s_prob[buf][lane][tt*2 + 1];
            const float* H = &s_hand[buf][lane][tt*6];
            const float* R = &s_rotm[buf][lane][tt*9];

            const int lcf = p0 > 0.5f;
            const int rcf = p1 > 0.5f;
            int l_c = 0, r_c = 0, l_s = 0, r_s = 0;
            if (t != 0) {                 // reference forces step 0 all-false
                l_c = lcf; r_c = rcf;
                l_s = lcf & (prev_lc ^ 1);
                r_s = rcf & (prev_rc ^ 1);
            }
            prev_lc = lcf; prev_rc = rcf;

            const int  nw      = l_s ? 0 : (r_s ? 1 : -1);
            const bool has_c   = (cur == 0 && l_c) || (cur == 1 && r_c);
            const bool has_o   = (cur == 0 && r_c) || (cur == 1 && l_c);
            const bool grab    = (nw != -1) || (!has_c && has_o);
            const int  curgrab = (nw != -1) ? nw : (1 - cur);
            const bool release = !grab && !has_c && (cur != -1);
            const int  newcur  = grab ? curgrab : (release ? -1 : cur);

            const int g01 = curgrab < 0 ? 0 : (curgrab > 1 ? 1 : curgrab);
            const float hx = H[g01*3 + 0], hy = H[g01*3 + 1], hz = H[g01*3 + 2];
            const float vx = px - hx, vy = py - hy, vz = pz - hz;
            const float d0 = sqrtf(vx*vx + vy*vy + vz*vz);
            const bool  valid = d0 > 1e-6f;
            const float inv = 1.0f / fmaxf(d0, 1e-6f);
            const float ux = valid ? vx * inv : 0.0f;
            const float uy = valid ? vy * inv : 0.0f;
            const float uz = valid ? vz * inv : 1.0f;
            const float dd = valid ? d0 : 0.1f;

            // dir_new = R^T * unit
            const float dnx = R[0]*ux + R[3]*uy + R[6]*uz;
            const float dny = R[1]*ux + R[4]*uy + R[7]*uz;
            const float dnz = R[2]*ux + R[5]*uy + R[8]*uz;
            if (grab) { dlx = dnx; dly = dny; dlz = dnz; dist = dd; }

            const bool attached = (newcur != -1);
            const int  a01 = newcur < 0 ? 0 : (newcur > 1 ? 1 : newcur);
            const float hcx = H[a01*3 + 0], hcy = H[a01*3 + 1], hcz = H[a01*3 + 2];
            // new_pos = hand + (R * dir_local) * dist
            const float nx = hcx + (R[0]*dlx + R[1]*dly + R[2]*dlz) * dist;
            const float ny = hcy + (R[3]*dlx + R[4]*dly + R[5]*dlz) * dist;
            const float nz = hcz + (R[6]*dlx + R[7]*dly + R[8]*dlz) * dist;
            if (attached) { px = nx; py = ny; pz = nz; }
            cur = newcur;

            if (live) {
                const u64 o = ((u64)b * T + t) * 3ull;
                out[o + 0] = px;
                out[o + 1] = py;
                out[o + 2] = pz;
            }
        }
    }
}

extern "C" void kernel_launch(void* const* d_in, const int* in_sizes, int n_in,
                              void* d_out, int out_size, void* d_ws, size_t ws_size,
                              hipStream_t stream) {
    (void)n_in; (void)d_ws; (void)ws_size; (void)out_size;
    const float* prob  = (const float*)d_in[0];   // [B][T][2]
    const float* hands = (const float*)d_in[1];   // [B][T][2][3]
    const float* rotm  = (const float*)d_in[2];   // [B][T][3][3]
    const float* tini  = (const float*)d_in[3];   // [B][3]
    float* out = (float*)d_out;                   // [B][T][3]

    const int B = in_sizes[3] / 3;
    const int T = in_sizes[0] / (2 * B);

    const int nblk = (B + LANES - 1) / LANES;
    fk_obj_trans_kernel<<<nblk, LANES, 0, stream>>>(prob, hands, rotm, tini, out, B, T);
}